// FeaturePropagation_30270929502479
// MI455X (gfx1250) — compile-verified
//
#include <hip/hip_runtime.h>
#include <stdint.h>

#ifndef N_ITER
#define N_ITER 40
#endif

constexpr int D      = 128;       // feature dim
constexpr int WPB    = 8;         // waves per block (wave32)
constexpr int TPB    = WPB * 32;  // 256 threads
constexpr int STAGES = 4;         // async pipeline depth (3 gathers in flight)

// ---------------- small utility kernels ----------------

__global__ void k_zero_f32(float* __restrict__ p, int n) {
  int i = blockIdx.x * blockDim.x + threadIdx.x;
  if (i < n) p[i] = 0.0f;
}

__global__ void k_zero_i32(int* __restrict__ p, int n) {
  int i = blockIdx.x * blockDim.x + threadIdx.x;
  if (i < n) p[i] = 0;
}

// deg[r] += 1 for every edge source r (normalization uses ALL edges)
__global__ void k_degree(const int* __restrict__ row, float* __restrict__ deg, int E) {
  int e = blockIdx.x * blockDim.x + threadIdx.x;
  if (e < E) atomicAdd(&deg[row[e]], 1.0f);
}

// deg -> dinv in place: dinv = deg > 0 ? rsqrt(deg) : 0
__global__ void k_rsqrt(float* __restrict__ deg, int n) {
  int i = blockIdx.x * blockDim.x + threadIdx.x;
  if (i < n) {
    float d = deg[i];
    deg[i] = (d > 0.0f) ? rsqrtf(d) : 0.0f;
  }
}

// Drop edges whose destination row is masked (those rows are pinned to x),
// and fold the per-edge weight in. Halves gather+atomic traffic for all iters.
__global__ void k_compact(const int* __restrict__ row, const int* __restrict__ col,
                          const float* __restrict__ dinv,
                          const unsigned char* __restrict__ mask,
                          int* __restrict__ row2, int* __restrict__ col2,
                          float* __restrict__ w2, int* __restrict__ cnt, int E) {
  int e = blockIdx.x * blockDim.x + threadIdx.x;
  if (e >= E) return;
  int r = row[e];
  if (mask[r]) return;
  int c = col[e];
  int p = atomicAdd(cnt, 1);
  row2[p] = r;
  col2[p] = c;
  w2[p]   = dinv[r] * dinv[c];
}

// Once per launch per buffer: out = mask ? x : 0  (float4 over nD/4 elements)
__global__ void k_init4(const float4* __restrict__ x4,
                        const unsigned char* __restrict__ mask,
                        float4* __restrict__ out4, int nD4) {
  int i = blockIdx.x * blockDim.x + threadIdx.x;
  if (i < nD4) {
    float4 z = make_float4(0.f, 0.f, 0.f, 0.f);
    out4[i] = mask[i >> 5] ? x4[i] : z;   // 32 float4 per 128-float row
  }
}

// Per iteration: zero only the unmasked rows (masked rows stay == x forever,
// since the compacted edge list never scatters into them).
__global__ void k_zero_unmasked4(const unsigned char* __restrict__ mask,
                                 float4* __restrict__ out4, int nD4) {
  int i = blockIdx.x * blockDim.x + threadIdx.x;
  if (i < nD4 && !mask[i >> 5])
    out4[i] = make_float4(0.f, 0.f, 0.f, 0.f);
}

// ---------------- main SpMM scatter kernel ----------------
// One wave per edge (grid-strided). The 128-float source row cur[col] is
// gathered with global_load_async_to_lds_b128 (32 lanes x 16B = 512B) into a
// 4-stage LDS ring, keeping 3 random L2 gathers in flight per wave while the
// oldest row is multiplied by w and scattered via global_atomic_add_f32.

__global__ __launch_bounds__(TPB) void k_scatter(
    const int* __restrict__ row2, const int* __restrict__ col2,
    const float* __restrict__ w2, const int* __restrict__ cntp,
    const float* __restrict__ cur, float* __restrict__ nxt) {
  __shared__ __align__(16) float smem[WPB][STAGES][D];  // 16 KB

  const int lane   = threadIdx.x & 31;
  const int wave   = threadIdx.x >> 5;
  const int gwave  = blockIdx.x * WPB + wave;
  const int nwaves = (int)gridDim.x * WPB;
  const int E2     = *cntp;

  // low 32 bits of the flat shared-aperture address == LDS byte address
  const uint32_t lds_base = (uint32_t)(uintptr_t)&smem[wave][0][lane * 4];
  // stage s lives at lds_base + s*512

  // prologue: fill STAGES-1 pipeline slots
  #pragma unroll
  for (int s = 0; s < STAGES - 1; ++s) {
    int ep = gwave + s * nwaves;
    if (ep < E2) {
      unsigned long long src =
          (unsigned long long)(uintptr_t)(cur + (size_t)col2[ep] * D + lane * 4);
      uint32_t ldsa = lds_base + (uint32_t)(s * (D * 4));
      asm volatile("global_load_async_to_lds_b128 %0, %1, off"
                   :: "v"(ldsa), "v"(src) : "memory");
    }
  }

  int buf = 0;
  for (int e = gwave; e < E2; e += nwaves) {
    int en = e + (STAGES - 1) * nwaves;
    if (en < E2) {
      unsigned long long srcn =
          (unsigned long long)(uintptr_t)(cur + (size_t)col2[en] * D + lane * 4);
      uint32_t ldsn = lds_base + (uint32_t)(((buf + STAGES - 1) & (STAGES - 1)) * (D * 4));
      asm volatile("global_load_async_to_lds_b128 %0, %1, off"
                   :: "v"(ldsn), "v"(srcn) : "memory");
      // in-order completion: <= STAGES-1 outstanding => oldest stage is ready
      asm volatile("s_wait_asynccnt 3" ::: "memory");
    } else {
      // tail (last <=3 edges of this wave): drain everything
      asm volatile("s_wait_asynccnt 0" ::: "memory");
    }

    float4 v  = *(const float4*)&smem[wave][buf][lane * 4];
    float  we = w2[e];
    float* dst = nxt + (size_t)row2[e] * D + lane * 4;
    atomicAdd(dst + 0, v.x * we);   // global_atomic_add_f32, resolves in L2
    atomicAdd(dst + 1, v.y * we);
    atomicAdd(dst + 2, v.z * we);
    atomicAdd(dst + 3, v.w * we);
    buf = (buf + 1) & (STAGES - 1);
  }
}

// ---------------- host side ----------------

extern "C" void kernel_launch(void* const* d_in, const int* in_sizes, int n_in,
                              void* d_out, int out_size, void* d_ws, size_t ws_size,
                              hipStream_t stream) {
  (void)n_in; (void)out_size; (void)ws_size;

  const float*         x    = (const float*)d_in[0];
  const int*           ei   = (const int*)d_in[1];
  const unsigned char* mask = (const unsigned char*)d_in[2];
  float*               out  = (float*)d_out;

  const int nD  = in_sizes[0];        // n * 128
  const int nD4 = nD / 4;
  const int n   = in_sizes[2];        // 100000
  const int E   = in_sizes[1] / 2;    // 1.6M
  const int* row = ei;
  const int* col = ei + E;

  // carve workspace (hipMalloc base is 256B aligned)
  uintptr_t p = (uintptr_t)d_ws;
  auto carve = [&](size_t bytes) -> void* {
    uintptr_t q = p;
    p = (p + bytes + 255) & ~(uintptr_t)255;
    return (void*)q;
  };
  float* deg  = (float*)carve(sizeof(float) * (size_t)n);   // then holds dinv
  int*   cnt  = (int*)  carve(sizeof(int));
  int*   row2 = (int*)  carve(sizeof(int)   * (size_t)E);
  int*   col2 = (int*)  carve(sizeof(int)   * (size_t)E);
  float* w2   = (float*)carve(sizeof(float) * (size_t)E);
  float* bufA = (float*)carve(sizeof(float) * (size_t)nD);

  const int tpb = 256;
  auto cdiv = [](int a, int b) { return (a + b - 1) / b; };

  // one-time preprocessing, amortized over 40 iterations
  k_zero_f32<<<cdiv(n, tpb), tpb, 0, stream>>>(deg, n);
  k_zero_i32<<<1, 64, 0, stream>>>(cnt, 1);
  k_degree  <<<cdiv(E, tpb), tpb, 0, stream>>>(row, deg, E);
  k_rsqrt   <<<cdiv(n, tpb), tpb, 0, stream>>>(deg, n);
  k_compact <<<cdiv(E, tpb), tpb, 0, stream>>>(row, col, deg, mask,
                                               row2, col2, w2, cnt, E);

  // Both ping-pong buffers: masked rows = x (stay that way all launch),
  // unmasked rows = 0. d_out holds X0; iteration 40 ends back in d_out.
  k_init4<<<cdiv(nD4, tpb), tpb, 0, stream>>>((const float4*)x, mask,
                                              (float4*)out, nD4);
  k_init4<<<cdiv(nD4, tpb), tpb, 0, stream>>>((const float4*)x, mask,
                                              (float4*)bufA, nD4);

  float* curb = out;
  float* nxtb = bufA;
  const int scatter_blocks = 1024;  // 8192 waves, ~98 compacted edges each
  for (int it = 0; it < N_ITER; ++it) {
    k_zero_unmasked4<<<cdiv(nD4, tpb), tpb, 0, stream>>>(mask, (float4*)nxtb, nD4);
    k_scatter<<<scatter_blocks, TPB, 0, stream>>>(row2, col2, w2, cnt, curb, nxtb);
    float* t = curb; curb = nxtb; nxtb = t;
  }
  // 40 swaps (even) => final result resides in d_out
}